// TemporalRunnerGNN_9079560864471
// MI455X (gfx1250) — compile-verified
//
#include <hip/hip_runtime.h>
#include <hip/hip_bf16.h>
#include <math.h>

// ---------------------------------------------------------------------------
// TemporalRunnerGNN forward for gfx1250 (MI455X), fp32, WMMA f32 16x16x4.
// ---------------------------------------------------------------------------

typedef float v2f __attribute__((ext_vector_type(2)));
typedef float v8f __attribute__((ext_vector_type(8)));

static constexpr int kP      = 23;    // nodes per frame graph
static constexpr int kEperG  = 506;   // 23*22 edges per graph
static constexpr int kInDeg  = 22;    // in-edges per node
static constexpr int kG      = 400;   // graphs
static constexpr int kB      = 8;
static constexpr int kT      = 50;
static constexpr int kN      = kG * kP;        // 9200 nodes
static constexpr int kE      = kG * kEperG;    // 202400 edges
static constexpr int kF0     = 16;    // NODE_FEAT
static constexpr int kD      = 64;    // per-head channels / model dim
static constexpr int kH      = 4;     // GNN heads
static constexpr int kHC     = kH * kD;  // 256

// ---------------------------------------------------------------------------
// edge_attr column std (ddof=1) -> per-column inverse scale in stats[8..11]
// ---------------------------------------------------------------------------
__global__ void ea_stats_init(float* stats) {
    if (blockIdx.x == 0 && threadIdx.x < 16) stats[threadIdx.x] = 0.0f;
}

__global__ void ea_stats(const float* __restrict__ ea, float* __restrict__ stats, int E) {
    float s[4]  = {0, 0, 0, 0};
    float ss[4] = {0, 0, 0, 0};
    for (int i = blockIdx.x * blockDim.x + threadIdx.x; i < E; i += gridDim.x * blockDim.x) {
        #pragma unroll
        for (int c = 0; c < 4; ++c) {
            float v = ea[(size_t)i * 4 + c];
            s[c] += v; ss[c] += v * v;
        }
    }
    #pragma unroll
    for (int c = 0; c < 4; ++c) {
        atomicAdd(&stats[c], s[c]);
        atomicAdd(&stats[4 + c], ss[c]);
    }
}

__global__ void ea_scale(float* stats, int E) {
    int c = threadIdx.x;
    if (blockIdx.x == 0 && c < 4) {
        float n    = (float)E;
        float mean = stats[c] / n;
        float var  = (stats[4 + c] - n * mean * mean) / (n - 1.0f);
        float sd   = sqrtf(fmaxf(var, 0.0f));
        sd         = fmaxf(sd, 1e-6f);
        stats[8 + c] = 1.0f / sd;
    }
}

// ---------------------------------------------------------------------------
// C[M,Nout] = A[M,K] @ W[K,Nout] + bias, via V_WMMA_F32_16X16X4_F32.
// 256-thread blocks = 8 waves; each wave owns a 16x64 output strip (4 N-tiles)
// so each A fragment load feeds 4 consecutive WMMAs (independent accumulators
// => no RAW hazard NOPs, dense matrix-pipe issue).
// Requirements: M%16==0, Nout%64==0 (here 256 and 64), K%4==0.
// Fragment layouts per CDNA5 ISA 7.12.2:
//   A (16x4 f32): lanes 0-15 hold M=lane, K={k,k+1}; lanes 16-31 K={k+2,k+3}
//   B (4x16 f32): lane = N column; lanes 0-15 rows k,k+1; lanes 16-31 k+2,k+3
//   C/D (16x16):  VGPR r -> M=r (lanes 0-15) / M=r+8 (lanes 16-31), N=lane%16
// ---------------------------------------------------------------------------
__global__ void gemm_f32_wmma(const float* __restrict__ A, const float* __restrict__ W,
                              const float* __restrict__ bias, float* __restrict__ C,
                              int M, int K, int Nout) {
    int wave = blockIdx.x * (blockDim.x >> 5) + (threadIdx.x >> 5);
    int lane = threadIdx.x & 31;
    int half = lane >> 4;            // 0 | 1
    int l    = lane & 15;

    int nGroups = Nout >> 6;         // groups of 4 N-tiles (64 columns)
    int mt = wave / nGroups;
    int ng = wave - mt * nGroups;
    if (mt >= (M >> 4)) return;      // wave-uniform exit; active waves keep EXEC=~0
    int m0 = mt << 4;
    int n0 = ng << 6;

    v8f acc0 = {0.f,0.f,0.f,0.f,0.f,0.f,0.f,0.f};
    v8f acc1 = acc0, acc2 = acc0, acc3 = acc0;

    const float* arow = A + (size_t)(m0 + l) * K + 2 * half;
    for (int k = 0; k < K; k += 4) {
        // prefetch next K-step of W (lowers to global_prefetch_b8)
        if (k + 4 < K)
            __builtin_prefetch(W + (size_t)(k + 4 + 2 * half) * Nout + n0 + l, 0, 1);

        v2f a; a.x = arow[k]; a.y = arow[k + 1];
        const float* bp = W + (size_t)(k + 2 * half) * Nout + n0 + l;
        v2f b0; b0.x = bp[0];  b0.y = bp[Nout];
        v2f b1; b1.x = bp[16]; b1.y = bp[Nout + 16];
        v2f b2; b2.x = bp[32]; b2.y = bp[Nout + 32];
        v2f b3; b3.x = bp[48]; b3.y = bp[Nout + 48];
        acc0 = __builtin_amdgcn_wmma_f32_16x16x4_f32(false, a, false, b0, (short)0, acc0, false, false);
        acc1 = __builtin_amdgcn_wmma_f32_16x16x4_f32(false, a, false, b1, (short)0, acc1, false, false);
        acc2 = __builtin_amdgcn_wmma_f32_16x16x4_f32(false, a, false, b2, (short)0, acc2, false, false);
        acc3 = __builtin_amdgcn_wmma_f32_16x16x4_f32(false, a, false, b3, (short)0, acc3, false, false);
    }

    float bv0 = bias ? bias[n0 + l]      : 0.0f;
    float bv1 = bias ? bias[n0 + l + 16] : 0.0f;
    float bv2 = bias ? bias[n0 + l + 32] : 0.0f;
    float bv3 = bias ? bias[n0 + l + 48] : 0.0f;
    #pragma unroll
    for (int r = 0; r < 8; ++r) {
        size_t row = (size_t)(m0 + r + 8 * half);
        float* crow = C + row * Nout + n0 + l;
        crow[0]  = acc0[r] + bv0;
        crow[16] = acc1[r] + bv1;
        crow[32] = acc2[r] + bv2;
        crow[48] = acc3[r] + bv3;
    }
}

// ---------------------------------------------------------------------------
// GNN edge attention: one wave per (graph g, dst node j).
// Complete digraph => dst j has 22 in-edges from all i != j; the reference's
// meshgrid edge order gives edge (s -> j) local index s*22 + (j<s ? j : j-1).
// Lanes 0..21 own one in-edge each for alpha/softmax; all 32 lanes own 2 of
// the 64 output channels for the aggregate. Heads looped; mean over heads +
// root skip + ReLU fused into the output.
// ---------------------------------------------------------------------------
__global__ void gnn_edge_attn(const float* __restrict__ Q, const float* __restrict__ K,
                              const float* __restrict__ V, const float* __restrict__ ea,
                              const float* __restrict__ We,     // [4, 256]
                              const float* __restrict__ scale,  // stats+8, 4 floats
                              const float* __restrict__ skip,   // [N, 64]
                              float* __restrict__ Hout) {
    int wave = (int)((blockIdx.x * blockDim.x + threadIdx.x) >> 5);
    int lane = threadIdx.x & 31;
    if (wave >= kN) return;                 // uniform per wave
    int g  = wave / kP;
    int j  = wave - g * kP;
    int jn = g * kP + j;

    float s0 = scale[0], s1 = scale[1], s2 = scale[2], s3 = scale[3];

    // lane i < 22 owns source node sl = (i<j) ? i : i+1
    int i  = lane;
    int sl = (i < j) ? i : i + 1;
    int sn = g * kP + sl;
    float ea0 = 0.f, ea1 = 0.f, ea2 = 0.f, ea3 = 0.f;
    if (i < kInDeg) {
        int lidx = sl * kInDeg + ((j < sl) ? j : (j - 1));
        const float* ep = ea + (size_t)(g * kEperG + lidx) * 4;
        ea0 = ep[0] * s0; ea1 = ep[1] * s1; ea2 = ep[2] * s2; ea3 = ep[3] * s3;
    }

    int da = lane;            // channels owned by this lane within a head
    int db = lane + 32;

    float hacc_a = 0.f, hacc_b = 0.f;
    for (int h = 0; h < kH; ++h) {
        const float* qrow = Q + (size_t)jn * kHC + h * kD;
        const float* Wc   = We + h * kD;               // We[c][h*64+d] = Wc[c*256 + d]

        float alpha = -1e30f;
        if (i < kInDeg) {
            const float* krow = K + (size_t)sn * kHC + h * kD;
            float acc = 0.f;
            for (int d = 0; d < kD; ++d) {
                float e = ea0 * Wc[d] + ea1 * Wc[256 + d] + ea2 * Wc[512 + d] + ea3 * Wc[768 + d];
                acc = fmaf(qrow[d], krow[d] + e, acc);
            }
            alpha = acc * 0.125f;                       // 1/sqrt(64)
        }
        // wave32 softmax over lanes 0..21
        float m = alpha;
        #pragma unroll
        for (int o = 16; o >= 1; o >>= 1) m = fmaxf(m, __shfl_xor(m, o, 32));
        float w = (i < kInDeg) ? __expf(alpha - m) : 0.0f;
        float s = w;
        #pragma unroll
        for (int o = 16; o >= 1; o >>= 1) s += __shfl_xor(s, o, 32);
        float winv = 1.0f / s;

        float agg_a = 0.f, agg_b = 0.f;
        for (int t = 0; t < kInDeg; ++t) {
            float wt  = __shfl(w, t, 32) * winv;
            int   snt = __shfl(sn, t, 32);
            float e0 = __shfl(ea0, t, 32), e1 = __shfl(ea1, t, 32);
            float e2 = __shfl(ea2, t, 32), e3 = __shfl(ea3, t, 32);
            const float* vrow = V + (size_t)snt * kHC + h * kD;
            float eda = e0 * Wc[da] + e1 * Wc[256 + da] + e2 * Wc[512 + da] + e3 * Wc[768 + da];
            float edb = e0 * Wc[db] + e1 * Wc[256 + db] + e2 * Wc[512 + db] + e3 * Wc[768 + db];
            agg_a = fmaf(wt, vrow[da] + eda, agg_a);
            agg_b = fmaf(wt, vrow[db] + edb, agg_b);
        }
        hacc_a += agg_a;
        hacc_b += agg_b;
    }
    // mean over heads + root skip + ReLU (applied after each conv in reference)
    float oa = hacc_a * 0.25f + skip[(size_t)jn * kD + da];
    float ob = hacc_b * 0.25f + skip[(size_t)jn * kD + db];
    Hout[(size_t)jn * kD + da] = fmaxf(oa, 0.0f);
    Hout[(size_t)jn * kD + db] = fmaxf(ob, 0.0f);
}

// ---------------------------------------------------------------------------
// Runner gather + positional encoding: xt[b,t,:] = H[g*23+runner_idx[g]] + pe[t]
// ---------------------------------------------------------------------------
__global__ void gather_pe(const float* __restrict__ H, const int* __restrict__ runner_idx,
                          const float* __restrict__ pe, float* __restrict__ xt) {
    int idx = blockIdx.x * blockDim.x + threadIdx.x;   // over G*64
    if (idx >= kG * kD) return;
    int g = idx >> 6;
    int d = idx & 63;
    int t = g % kT;
    int node = g * kP + runner_idx[g];
    xt[idx] = H[(size_t)node * kD + d] + pe[t * kD + d];
}

// ---------------------------------------------------------------------------
// One nn.TransformerEncoderLayer (post-LN, relu, 4 heads, causal+pad mask),
// one workgroup (256 threads) per batch row, fully LDS-resident.
// LDS: Xs[50*64] + Ss[12800] = 64KB. Ss holds QKV [50*192] then attn-out o
// at +9600 [50*64]; after LN1 the whole Ss is reused for the FF hidden.
// ---------------------------------------------------------------------------
__global__ void encoder_layer(float* __restrict__ xt, const int* __restrict__ lengths,
                              const float* __restrict__ Wqkv, const float* __restrict__ bqkv,
                              const float* __restrict__ Wo,   const float* __restrict__ bo,
                              const float* __restrict__ W1,   const float* __restrict__ b1,
                              const float* __restrict__ W2,   const float* __restrict__ b2,
                              const float* __restrict__ ln1s, const float* __restrict__ ln1b,
                              const float* __restrict__ ln2s, const float* __restrict__ ln2b) {
    __shared__ float Xs[kT * kD];     // 3200
    __shared__ float Ss[12800];       // qkv / o / hidden

    int b   = blockIdx.x;
    int tid = threadIdx.x;
    int len = lengths[b];
    float* xg = xt + (size_t)b * kT * kD;

    for (int idx = tid; idx < kT * kD; idx += blockDim.x) Xs[idx] = xg[idx];
    __syncthreads();

    // --- QKV projection: Ss[t*192 + c] ---
    for (int idx = tid; idx < kT * 192; idx += blockDim.x) {
        int t = idx / 192, c = idx - t * 192;
        const float* xr = Xs + t * kD;
        float acc = bqkv[c];
        for (int k2 = 0; k2 < kD; ++k2) acc = fmaf(xr[k2], Wqkv[k2 * 192 + c], acc);
        Ss[idx] = acc;
    }
    __syncthreads();

    // --- attention, one thread per (head, query) row; online softmax ---
    if (tid < kH * kT) {
        int h  = tid / kT, tq = tid - h * kT;
        const float* qr = Ss + tq * 192 + h * 16;
        float m = -1e30f, s = 0.0f, acc[16];
        #pragma unroll
        for (int d = 0; d < 16; ++d) acc[d] = 0.0f;
        int kmax = (tq + 1 < len) ? (tq + 1) : len;    // causal AND pad masking
        for (int tk = 0; tk < kmax; ++tk) {
            const float* kr = Ss + tk * 192 + 64 + h * 16;
            float sc = 0.0f;
            #pragma unroll
            for (int d = 0; d < 16; ++d) sc = fmaf(qr[d], kr[d], sc);
            sc *= 0.25f;                               // 1/sqrt(16)
            float mn   = fmaxf(m, sc);
            float corr = __expf(m - mn);
            float w    = __expf(sc - mn);
            s = s * corr + w;
            const float* vr = Ss + tk * 192 + 128 + h * 16;
            #pragma unroll
            for (int d = 0; d < 16; ++d) acc[d] = fmaf(w, vr[d], acc[d] * corr);
            m = mn;
        }
        float inv = 1.0f / s;
        float* orow = Ss + 9600 + tq * kD + h * 16;
        #pragma unroll
        for (int d = 0; d < 16; ++d) orow[d] = acc[d] * inv;
    }
    __syncthreads();

    // --- o @ Wo + residual + LN1, one thread per row ---
    if (tid < kT) {
        int t = tid;
        const float* orow = Ss + 9600 + t * kD;
        float y[kD];
        float mu = 0.0f;
        for (int d = 0; d < kD; ++d) {
            float acc = bo[d];
            for (int k2 = 0; k2 < kD; ++k2) acc = fmaf(orow[k2], Wo[k2 * kD + d], acc);
            acc += Xs[t * kD + d];
            y[d] = acc; mu += acc;
        }
        mu *= (1.0f / kD);
        float var = 0.0f;
        for (int d = 0; d < kD; ++d) { float c = y[d] - mu; var = fmaf(c, c, var); }
        var *= (1.0f / kD);
        float rs = rsqrtf(var + 1e-5f);
        for (int d = 0; d < kD; ++d) Xs[t * kD + d] = (y[d] - mu) * rs * ln1s[d] + ln1b[d];
    }
    __syncthreads();

    // --- FF hidden = relu(X @ W1 + b1) into full Ss [50*256] ---
    for (int idx = tid; idx < kT * 256; idx += blockDim.x) {
        int t = idx >> 8, c = idx & 255;
        const float* xr = Xs + t * kD;
        float acc = b1[c];
        for (int k2 = 0; k2 < kD; ++k2) acc = fmaf(xr[k2], W1[k2 * 256 + c], acc);
        Ss[idx] = fmaxf(acc, 0.0f);
    }
    __syncthreads();

    // --- hidden @ W2 + residual + LN2 -> global, one thread per row ---
    if (tid < kT) {
        int t = tid;
        const float* hr = Ss + t * 256;
        float y[kD];
        float mu = 0.0f;
        for (int d = 0; d < kD; ++d) {
            float acc = b2[d];
            for (int k2 = 0; k2 < 256; ++k2) acc = fmaf(hr[k2], W2[k2 * kD + d], acc);
            acc += Xs[t * kD + d];
            y[d] = acc; mu += acc;
        }
        mu *= (1.0f / kD);
        float var = 0.0f;
        for (int d = 0; d < kD; ++d) { float c = y[d] - mu; var = fmaf(c, c, var); }
        var *= (1.0f / kD);
        float rs = rsqrtf(var + 1e-5f);
        for (int d = 0; d < kD; ++d) xg[t * kD + d] = (y[d] - mu) * rs * ln2s[d] + ln2b[d];
    }
}

// ---------------------------------------------------------------------------
// Final projection: out[b,t,:] = xt[b,t,:] @ Wp + bp   (800 outputs)
// ---------------------------------------------------------------------------
__global__ void proj_head(const float* __restrict__ xt, const float* __restrict__ Wp,
                          const float* __restrict__ bp, float* __restrict__ out) {
    int idx = blockIdx.x * blockDim.x + threadIdx.x;   // over B*T*2
    if (idx >= kB * kT * 2) return;
    int row = idx >> 1, c = idx & 1;
    const float* xr = xt + (size_t)row * kD;
    float acc = bp[c];
    for (int k2 = 0; k2 < kD; ++k2) acc = fmaf(xr[k2], Wp[k2 * 2 + c], acc);
    out[idx] = acc;
}

// ---------------------------------------------------------------------------
// kernel_launch
// Input order = setup_inputs() insertion order, params pytree flattened in
// insertion order:
//   0 x[9200,16] 1 edge_attr[202400,4] 2 src 3 dst 4 runner_idx[400] 5 lengths[8]
//   g1: 6 Wq[16,256] 7 bq 8 Wk 9 bk 10 Wv 11 bv 12 We[4,256] 13 Ws[16,64] 14 bs
//   g2: 15..23 (same, fin=64)
//   24 pe[1,100,64]
//   layer l (p = 25 + 12*l): p+0 Wqkv[64,192] p+1 bqkv p+2 Wo[64,64] p+3 bo
//       p+4 W1[64,256] p+5 b1 p+6 W2[256,64] p+7 b2
//       p+8 ln1_s p+9 ln1_b p+10 ln2_s p+11 ln2_b
//   49 Wp[64,2] 50 bp[2]
// ---------------------------------------------------------------------------
extern "C" void kernel_launch(void* const* d_in, const int* in_sizes, int n_in,
                              void* d_out, int out_size, void* d_ws, size_t ws_size,
                              hipStream_t stream) {
    (void)in_sizes; (void)n_in; (void)out_size; (void)ws_size;

    auto F = [&](int i) { return (const float*)d_in[i]; };
    const float* x          = F(0);
    const float* ea         = F(1);
    const int*   runner_idx = (const int*)d_in[4];
    const int*   lengths    = (const int*)d_in[5];

    // workspace layout (floats)
    float* ws    = (float*)d_ws;
    float* stats = ws;                       // 16
    float* Qb    = ws + 16;                  // 9200*256
    float* Kb    = Qb + (size_t)kN * kHC;
    float* Vb    = Kb + (size_t)kN * kHC;
    float* skipb = Vb + (size_t)kN * kHC;    // 9200*64
    float* H1    = skipb + (size_t)kN * kD;
    float* H2    = H1 + (size_t)kN * kD;
    float* xtb   = H2 + (size_t)kN * kD;     // 8*50*64

    // 1) edge_attr column std -> inverse scales
    ea_stats_init<<<1, 32, 0, stream>>>(stats);
    ea_stats<<<256, 256, 0, stream>>>(ea, stats, kE);
    ea_scale<<<1, 32, 0, stream>>>(stats, kE);

    const int mtiles = kN / 16;                           // 575 row tiles
    auto gemmBlocks = [&](int Nout) {                     // 8 waves per block
        int waves = mtiles * (Nout >> 6);
        return (waves + 7) / 8;
    };

    // 2) layer-1 node linears (K=16) via f32 WMMA
    gemm_f32_wmma<<<gemmBlocks(kHC), 256, 0, stream>>>(x, F(6),  F(7),  Qb,    kN, kF0, kHC);
    gemm_f32_wmma<<<gemmBlocks(kHC), 256, 0, stream>>>(x, F(8),  F(9),  Kb,    kN, kF0, kHC);
    gemm_f32_wmma<<<gemmBlocks(kHC), 256, 0, stream>>>(x, F(10), F(11), Vb,    kN, kF0, kHC);
    gemm_f32_wmma<<<gemmBlocks(kD),  256, 0, stream>>>(x, F(13), F(14), skipb, kN, kF0, kD);

    // 3) layer-1 edge attention + mean-head + skip + relu
    gnn_edge_attn<<<(kN * 32) / 256, 256, 0, stream>>>(Qb, Kb, Vb, ea, F(12),
                                                       stats + 8, skipb, H1);

    // 4) layer-2 node linears (K=64)
    gemm_f32_wmma<<<gemmBlocks(kHC), 256, 0, stream>>>(H1, F(15), F(16), Qb,    kN, kD, kHC);
    gemm_f32_wmma<<<gemmBlocks(kHC), 256, 0, stream>>>(H1, F(17), F(18), Kb,    kN, kD, kHC);
    gemm_f32_wmma<<<gemmBlocks(kHC), 256, 0, stream>>>(H1, F(19), F(20), Vb,    kN, kD, kHC);
    gemm_f32_wmma<<<gemmBlocks(kD),  256, 0, stream>>>(H1, F(22), F(23), skipb, kN, kD, kD);

    // 5) layer-2 edge attention
    gnn_edge_attn<<<(kN * 32) / 256, 256, 0, stream>>>(Qb, Kb, Vb, ea, F(21),
                                                       stats + 8, skipb, H2);

    // 6) runner gather + positional encoding
    gather_pe<<<(kG * kD + 255) / 256, 256, 0, stream>>>(H2, runner_idx, F(24), xtb);

    // 7) temporal encoder, 2 layers (ln order: s then b, per enc_params insertion)
    for (int l = 0; l < 2; ++l) {
        int p = 25 + l * 12;
        encoder_layer<<<kB, 256, 0, stream>>>(xtb, lengths,
                                              F(p + 0), F(p + 1), F(p + 2), F(p + 3),
                                              F(p + 4), F(p + 5), F(p + 6), F(p + 7),
                                              F(p + 8), F(p + 9),    // ln1_s, ln1_b
                                              F(p + 10), F(p + 11)); // ln2_s, ln2_b
    }

    // 8) projection head -> d_out [8,50,2]
    proj_head<<<(kB * kT * 2 + 255) / 256, 256, 0, stream>>>(xtb, F(49), F(50), (float*)d_out);
}